// OptimNet_72533407695237
// MI455X (gfx1250) — compile-verified
//
#include <hip/hip_runtime.h>

#define S_DIM 2048
#define B_DIM 16

typedef float v2f __attribute__((ext_vector_type(2)));
typedef float v8f __attribute__((ext_vector_type(8)));
typedef unsigned int u32x4 __attribute__((ext_vector_type(4)));
typedef int i32x4 __attribute__((ext_vector_type(4)));
typedef int i32x8 __attribute__((ext_vector_type(8)));

#define KC 256           // K-chunk (floats) staged per TDM transfer
#define LDS_STRIDE 260   // 256 + 4 pad floats -> bank-conflict-free column reads

// ---------------------------------------------------------------------------
// Tensor Data Mover: async 2D tile load Global -> LDS.
//   rows x KC f32 tile, global row stride = stride_elems, LDS rows padded to
//   LDS_STRIDE via the TDM pad feature (pad 4 DWORDs every 256 DWORDs).
// Tracked on TENSORcnt; issued by one wave, completion fanned out via barrier.
// ---------------------------------------------------------------------------
__device__ __forceinline__ void tdm_load_2d(unsigned int lds_off, const float* gptr,
                                            int rows, int cols, int stride_elems) {
  unsigned long long ga = (unsigned long long)(uintptr_t)gptr;
  u32x4 g0;
  g0[0] = 1u;                                        // count=1, user mode, no gather
  g0[1] = lds_off;                                   // LDS byte address
  g0[2] = (unsigned int)(ga & 0xffffffffu);          // global_addr[31:0]
  g0[3] = (unsigned int)((ga >> 32) & 0x1ffffffu)    // global_addr[56:32]
          | (2u << 30);                              // type = 2 ("image")
  const unsigned int dim0 = 0x00100000u;             // huge tensor dims: no OOB clamp
  const unsigned int dim1 = 0x00100000u;
  i32x8 g1;
  g1[0] = (int)((2u << 16)                           // data_size = 4 bytes
                | (1u << 20)                         // pad_enable
                | (7u << 22)                         // pad_interval: 256 DWORDs
                | (3u << 25));                       // pad_amount:   4 DWORDs
  g1[1] = (int)((dim0 & 0xffffu) << 16);             // tensor_dim0[15:0]
  g1[2] = (int)(((dim0 >> 16) & 0xffffu) | ((dim1 & 0xffffu) << 16));
  g1[3] = (int)(((dim1 >> 16) & 0xffffu) | (((unsigned)cols & 0xffffu) << 16)); // tile_dim0
  g1[4] = (int)((unsigned)rows & 0xffffu);           // tile_dim1 (tile_dim2 = 0)
  g1[5] = (int)(unsigned)stride_elems;               // tensor_dim0_stride[31:0]
  g1[6] = 0;
  g1[7] = 0;
  i32x4 z4 = {0, 0, 0, 0};
#if defined(__clang_major__) && __clang_major__ >= 23
  i32x8 z8 = {0, 0, 0, 0, 0, 0, 0, 0};
  __builtin_amdgcn_tensor_load_to_lds(g0, g1, z4, z4, z8, 0);
#else
  __builtin_amdgcn_tensor_load_to_lds(g0, g1, z4, z4, 0);
#endif
}

// ---------------------------------------------------------------------------
// GEMM: Y(16 x N) = X(16 x K) @ W(N x K)^T + bias   (row-major, f32)
// One 256-thread block per 16-column tile. Double-buffered TDM staging of the
// 16x256 X and W tiles into LDS; 8 waves split K within each chunk (8 WMMAs
// per wave per chunk) and run V_WMMA_F32_16X16X4_F32 out of LDS; partial
// tiles reduced deterministically through LDS.
// ---------------------------------------------------------------------------
__global__ __launch_bounds__(256)
void gemm16_wmma(const float* __restrict__ X, const float* __restrict__ W,
                 const float* __restrict__ bias, float* __restrict__ Y,
                 int K, int N) {
  __shared__ float lds_x[2][16][LDS_STRIDE];
  __shared__ float lds_w[2][16][LDS_STRIDE];
  __shared__ float red[8 * 256];

  const int tid  = threadIdx.x;
  const int wv   = tid >> 5;
  const int lane = tid & 31;
  const int n0   = blockIdx.x * 16;
  const int m    = lane & 15;          // A row (M) / B column (N) within tile
  const int koff = (lane >> 4) * 2;    // lane-half selects K offset 0 or 2
  int rows = N - n0; rows = rows > 16 ? 16 : rows;   // W tail clamp (out=1 block)
  const float* Wn = W + (size_t)n0 * K;
  const bool lead = (wv == 0);

  v8f acc = {0.f, 0.f, 0.f, 0.f, 0.f, 0.f, 0.f, 0.f};
  const int NCH = K >> 8;              // K / 256 (K == 2048 here -> 8 chunks)

  if (lead) {
    tdm_load_2d((unsigned int)(uintptr_t)&lds_x[0][0][0], X, 16, KC, K);
    tdm_load_2d((unsigned int)(uintptr_t)&lds_w[0][0][0], Wn, rows, KC, K);
  }
  for (int c = 0; c < NCH; ++c) {
    const int buf = c & 1;
    if (lead) {
      if (c + 1 < NCH) {
        const int k1 = (c + 1) << 8;
        tdm_load_2d((unsigned int)(uintptr_t)&lds_x[buf ^ 1][0][0], X + k1, 16, KC, K);
        tdm_load_2d((unsigned int)(uintptr_t)&lds_w[buf ^ 1][0][0], Wn + k1, rows, KC, K);
        __builtin_amdgcn_s_wait_tensorcnt(2);   // chunk c's two DMAs are complete
      } else {
        __builtin_amdgcn_s_wait_tensorcnt(0);   // drain
      }
    }
    __syncthreads();                            // tile visible to all waves
    const float (*lx)[LDS_STRIDE] = lds_x[buf];
    const float (*lw)[LDS_STRIDE] = lds_w[buf];
#pragma unroll
    for (int t = 0; t < 8; ++t) {
      const int kk = (wv << 5) + (t << 2) + koff;
      v2f a = *(const v2f*)&lx[m][kk];
      v2f b = *(const v2f*)&lw[m][kk];
      acc = __builtin_amdgcn_wmma_f32_16x16x4_f32(false, a, false, b,
                                                  (short)0, acc, false, false);
    }
    __syncthreads();                            // done before buffer is re-filled
  }
#pragma unroll
  for (int r = 0; r < 8; ++r) red[wv * 256 + lane * 8 + r] = acc[r];
  __syncthreads();
  float sum = 0.f;
#pragma unroll
  for (int w = 0; w < 8; ++w) sum += red[w * 256 + tid];
  // map flat slot -> (lane, reg) -> (M, N) per C/D layout
  const int l = tid >> 3, r = tid & 7;
  const int mm = ((l >> 4) << 3) + r;
  const int nn = n0 + (l & 15);
  if (nn < N) Y[(size_t)mm * N + nn] = sum + (bias ? bias[nn] : 0.f);
}

// ---------------------------------------------------------------------------
// Per-batch global max of the outer-product scores:
//   m_b = max over {qmax,qmin} x {kmax,kmin} products
// ---------------------------------------------------------------------------
__global__ __launch_bounds__(256)
void attn_minmax(const float* __restrict__ q, const float* __restrict__ k,
                 float* __restrict__ mbuf) {
  const int b = blockIdx.x, tid = threadIdx.x;
  __shared__ float rqmn[256], rqmx[256], rkmn[256], rkmx[256];
  float qmn = 3.4e38f, qmx = -3.4e38f, kmn = 3.4e38f, kmx = -3.4e38f;
  for (int j = tid; j < S_DIM; j += 256) {
    float x = q[b * S_DIM + j];
    float y = k[b * S_DIM + j];
    qmn = fminf(qmn, x); qmx = fmaxf(qmx, x);
    kmn = fminf(kmn, y); kmx = fmaxf(kmx, y);
  }
  rqmn[tid] = qmn; rqmx[tid] = qmx; rkmn[tid] = kmn; rkmx[tid] = kmx;
  __syncthreads();
  for (int st = 128; st; st >>= 1) {
    if (tid < st) {
      rqmn[tid] = fminf(rqmn[tid], rqmn[tid + st]);
      rqmx[tid] = fmaxf(rqmx[tid], rqmx[tid + st]);
      rkmn[tid] = fminf(rkmn[tid], rkmn[tid + st]);
      rkmx[tid] = fmaxf(rkmx[tid], rkmx[tid + st]);
    }
    __syncthreads();
  }
  if (tid == 0) {
    float a = rqmx[0] * rkmx[0], c = rqmx[0] * rkmn[0];
    float d = rqmn[0] * rkmx[0], e = rqmn[0] * rkmn[0];
    mbuf[b] = fmaxf(fmaxf(a, c), fmaxf(d, e));
  }
}

// ---------------------------------------------------------------------------
// Per-row streaming softmax partials. Each block: 8 rows of one batch; k[b,:]
// and v[b,:] are staged once in LDS (8x less L2 traffic), one wave32 per row.
//   s[b,i]  = sum_j exp(q_i k_j - m_b)
//   sv[b,i] = sum_j exp(q_i k_j - m_b) * v_j
// ---------------------------------------------------------------------------
__global__ __launch_bounds__(256)
void attn_rows(const float* __restrict__ q, const float* __restrict__ k,
               const float* __restrict__ v, const float* __restrict__ mbuf,
               float* __restrict__ s, float* __restrict__ sv) {
  __shared__ float lk[S_DIM], lv[S_DIM];
  const int tid = threadIdx.x;
  const int b   = blockIdx.x >> 8;            // 256 blocks per batch (2048 / 8)
  const int i0  = (blockIdx.x & 255) * 8;
  for (int j = tid; j < S_DIM; j += 256) {
    lk[j] = k[(size_t)b * S_DIM + j];
    lv[j] = v[(size_t)b * S_DIM + j];
  }
  __syncthreads();
  const int wv = tid >> 5, lane = tid & 31;
  const int i  = i0 + wv;
  const float qi = q[(size_t)b * S_DIM + i];
  const float m  = mbuf[b];
  float as = 0.f, av = 0.f;
  for (int j = lane; j < S_DIM; j += 32) {
    float e = __expf(qi * lk[j] - m);
    as += e;
    av += e * lv[j];
  }
#pragma unroll
  for (int off = 16; off; off >>= 1) {
    as += __shfl_xor(as, off, 32);
    av += __shfl_xor(av, off, 32);
  }
  if (lane == 0) {
    s[(size_t)b * S_DIM + i]  = as;
    sv[(size_t)b * S_DIM + i] = av;
  }
}

// ---------------------------------------------------------------------------
// Finalize attention (Z = sum_i s_i, valued = sv/Z) + 3-term activation mix
// with softmax(act) weights:  fc = sigmoid(x)*x*p0 + sin(x)*p1 + x*p2
// ---------------------------------------------------------------------------
__global__ __launch_bounds__(256)
void attn_fc(const float* __restrict__ s, const float* __restrict__ sv,
             const float* __restrict__ act, float* __restrict__ fc) {
  const int b = blockIdx.x, tid = threadIdx.x;
  __shared__ float red[256];
  float z = 0.f;
  for (int j = tid; j < S_DIM; j += 256) z += s[(size_t)b * S_DIM + j];
  red[tid] = z; __syncthreads();
  for (int st = 128; st; st >>= 1) {
    if (tid < st) red[tid] += red[tid + st];
    __syncthreads();
  }
  const float Z = red[0];
  float a0 = act[0], a1 = act[1], a2 = act[2], a3 = act[3], a4 = act[4];
  float mx = fmaxf(fmaxf(fmaxf(a0, a1), fmaxf(a2, a3)), a4);
  float e0 = __expf(a0 - mx), e1 = __expf(a1 - mx), e2 = __expf(a2 - mx);
  float e3 = __expf(a3 - mx), e4 = __expf(a4 - mx);
  float inv = 1.f / (e0 + e1 + e2 + e3 + e4);
  float p0 = e0 * inv, p1 = e1 * inv, p2 = e2 * inv;
  for (int j = tid; j < S_DIM; j += 256) {
    float val = sv[(size_t)b * S_DIM + j] / Z;
    float sig = 1.f / (1.f + __expf(-val));
    fc[(size_t)b * S_DIM + j] = sig * val * p0 + sinf(val) * p1 + val * p2;
  }
}

// ---------------------------------------------------------------------------
// Row LayerNorm: one block per batch row, N up to 2048 (N==1 degenerates to be)
// ---------------------------------------------------------------------------
__global__ __launch_bounds__(256)
void ln16(const float* __restrict__ X, const float* __restrict__ g,
          const float* __restrict__ be, float* __restrict__ Y, int N) {
  const int b = blockIdx.x, tid = threadIdx.x;
  __shared__ float rs[256], rq[256];
  float sm = 0.f, sq = 0.f;
  for (int j = tid; j < N; j += 256) {
    float x = X[(size_t)b * N + j];
    sm += x; sq += x * x;
  }
  rs[tid] = sm; rq[tid] = sq; __syncthreads();
  for (int st = 128; st; st >>= 1) {
    if (tid < st) { rs[tid] += rs[tid + st]; rq[tid] += rq[tid + st]; }
    __syncthreads();
  }
  const float mu = rs[0] / (float)N;
  float var = rq[0] / (float)N - mu * mu;
  var = fmaxf(var, 0.f);
  const float rstd = rsqrtf(var + 1e-5f);
  for (int j = tid; j < N; j += 256) {
    float x = X[(size_t)b * N + j];
    Y[(size_t)b * N + j] = (x - mu) * rstd * g[j] + be[j];
  }
}

__global__ __launch_bounds__(256)
void add_hidden(const float* __restrict__ h1, const float* __restrict__ hid,
                float* __restrict__ o) {
  int i = blockIdx.x * 256 + threadIdx.x;
  if (i < B_DIM * S_DIM) o[i] = h1[i] + hid[i & (S_DIM - 1)];
}

__global__ __launch_bounds__(256)
void sigmoid_if(float* __restrict__ x, int n, const int* __restrict__ flag) {
  int i = blockIdx.x * 256 + threadIdx.x;
  if (i < n && flag[0] != 0) x[i] = 1.f / (1.f + __expf(-x[i]));
}

// ---------------------------------------------------------------------------
// Host-side orchestration
// ---------------------------------------------------------------------------
enum {
  OFF_Q  = 0,
  OFF_K  = 32768,
  OFF_V  = 65536,
  OFF_S  = 98304,
  OFF_SV = 131072,
  OFF_FC = 163840,
  OFF_T1 = 196608,
  OFF_T2 = 229376,
  OFF_U  = 262144,
  OFF_H1 = 294912,
  OFF_X2 = 327680,
  OFF_MB = 360448
};

static void run_attn_block(const float* x, void* const* d_in, int pbase,
                           float* out, int Nout, float* ws, hipStream_t stream) {
  const float* wq  = (const float*)d_in[pbase + 0];
  const float* wk  = (const float*)d_in[pbase + 1];
  const float* wv  = (const float*)d_in[pbase + 2];
  const float* act = (const float*)d_in[pbase + 3];
  const float* w1  = (const float*)d_in[pbase + 4];
  const float* b1  = (const float*)d_in[pbase + 5];
  const float* g1  = (const float*)d_in[pbase + 6];
  const float* be1 = (const float*)d_in[pbase + 7];
  const float* w2  = (const float*)d_in[pbase + 8];
  const float* b2  = (const float*)d_in[pbase + 9];
  const float* g2  = (const float*)d_in[pbase + 10];
  const float* be2 = (const float*)d_in[pbase + 11];

  float* Q  = ws + OFF_Q;
  float* K  = ws + OFF_K;
  float* V  = ws + OFF_V;
  float* Sb = ws + OFF_S;
  float* SV = ws + OFF_SV;
  float* FC = ws + OFF_FC;
  float* T1 = ws + OFF_T1;
  float* T2 = ws + OFF_T2;
  float* U  = ws + OFF_U;
  float* MB = ws + OFF_MB;

  const int tiles = S_DIM / 16;                 // 128 tiles per full GEMM
  gemm16_wmma<<<tiles, 256, 0, stream>>>(x, wq, nullptr, Q, S_DIM, S_DIM);
  gemm16_wmma<<<tiles, 256, 0, stream>>>(x, wk, nullptr, K, S_DIM, S_DIM);
  gemm16_wmma<<<tiles, 256, 0, stream>>>(x, wv, nullptr, V, S_DIM, S_DIM);
  attn_minmax<<<B_DIM, 256, 0, stream>>>(Q, K, MB);
  attn_rows<<<(B_DIM * S_DIM) / 8, 256, 0, stream>>>(Q, K, V, MB, Sb, SV);
  attn_fc<<<B_DIM, 256, 0, stream>>>(Sb, SV, act, FC);
  gemm16_wmma<<<tiles, 256, 0, stream>>>(FC, w1, b1, T1, S_DIM, S_DIM);
  ln16<<<B_DIM, 256, 0, stream>>>(T1, g1, be1, T2, S_DIM);
  gemm16_wmma<<<(Nout + 15) / 16, 256, 0, stream>>>(T2, w2, b2, U, S_DIM, Nout);
  ln16<<<B_DIM, 256, 0, stream>>>(U, g2, be2, out, Nout);
}

extern "C" void kernel_launch(void* const* d_in, const int* in_sizes, int n_in,
                              void* d_out, int out_size, void* d_ws, size_t ws_size,
                              hipStream_t stream) {
  (void)in_sizes; (void)n_in; (void)out_size; (void)ws_size;
  const float* p_grad = (const float*)d_in[0];
  const float* hidden = (const float*)d_in[1];
  const int*   is_forget = (const int*)d_in[50];

  float* out_a = (float*)d_out;                 // (16, 1024)
  float* out_c = out_a + B_DIM * 1024;          // (16, 1)
  float* out_h = out_c + B_DIM;                 // (16, 2048)
  float* ws = (float*)d_ws;
  float* H1 = ws + OFF_H1;
  float* X2 = ws + OFF_X2;

  // h1 = attn(p_grad, ph)
  run_attn_block(p_grad, d_in, 2, H1, S_DIM, ws, stream);
  // x2 = h1 + hidden (broadcast)
  add_hidden<<<(B_DIM * S_DIM + 255) / 256, 256, 0, stream>>>(H1, hidden, X2);
  // h = attn(x2, ph2)  -> directly into output slot
  run_attn_block(X2, d_in, 14, out_h, S_DIM, ws, stream);
  // c = attn(h, pc)    (out dim 1)
  run_attn_block(out_h, d_in, 26, out_c, 1, ws, stream);
  // a = attn(h, pa)    (out dim 1024), then optional sigmoid
  run_attn_block(out_h, d_in, 38, out_a, 1024, ws, stream);
  sigmoid_if<<<(B_DIM * 1024 + 255) / 256, 256, 0, stream>>>(out_a, B_DIM * 1024, is_forget);
}